// GNNLayer_30975304139246
// MI455X (gfx1250) — compile-verified
//
#include <hip/hip_runtime.h>
#include <hip/hip_bf16.h>
#include <math.h>

typedef float v2f __attribute__((ext_vector_type(2)));
typedef float v8f __attribute__((ext_vector_type(8)));

#define Bx 32
#define Nx 128
#define Dx 64
#define Hx 128
#define BN (Bx * Nx) /* 4096 rows */

__device__ __forceinline__ v8f wmma4(v2f a, v2f b, v8f c) {
  // D = A(16x4 f32) * B(4x16 f32) + C(16x16 f32)
  return __builtin_amdgcn_wmma_f32_16x16x4_f32(false, a, false, b, (short)0, c,
                                               false, false);
}

// ---------------------------------------------------------------------------
// Kernel 1: projections.  P[0]=x@W1[:64], P[1]=x@W1[64:], P[2]=x@A1[:64],
// P[3]=x@A1[64:]  -- each (4096 x 128).  grid 256 blocks (one 16-row M tile),
// 8 waves/block, each wave owns 4 of the 32 (proj, ntile) output tiles and
// reuses the same A fragment for all of them.
// ---------------------------------------------------------------------------
__global__ void __launch_bounds__(256)
proj_kernel(const float* __restrict__ x, const float* __restrict__ W1,
            const float* __restrict__ A1, float* __restrict__ P) {
  const int lane = threadIdx.x & 31;
  const int wave = threadIdx.x >> 5;
  const int mbase = blockIdx.x * 16;
  const int row = lane & 15;   // M row (A) / N col (B)
  const int kh = lane >> 4;    // K half select

  const float* xrow = x + (size_t)(mbase + row) * Dx;

  const float* wptr[4];
  int ncol[4];
  int projq[4];
  for (int q = 0; q < 4; ++q) {
    int tid = wave * 4 + q;       // 0..31
    int proj = tid >> 3;          // 0..3
    int nt = tid & 7;             // 0..7
    const float* Wsel = (proj < 2) ? W1 : A1;
    wptr[q] = Wsel + (size_t)((proj & 1) * 64) * Hx;
    ncol[q] = nt * 16;
    projq[q] = proj;
  }

  v8f acc[4] = {};
  for (int ks = 0; ks < Dx / 4; ++ks) {
    const int k0 = ks * 4 + kh * 2;
    v2f a;
    a.x = xrow[k0];
    a.y = xrow[k0 + 1];
    for (int q = 0; q < 4; ++q) {
      const int col = ncol[q] + row;
      v2f bf;
      bf.x = wptr[q][(size_t)k0 * Hx + col];
      bf.y = wptr[q][(size_t)(k0 + 1) * Hx + col];
      acc[q] = wmma4(a, bf, acc[q]);
    }
  }

  for (int q = 0; q < 4; ++q) {
    float* outp = P + (size_t)projq[q] * BN * Hx;
    const int col = ncol[q] + row;
    for (int v = 0; v < 8; ++v) {
      const int r = mbase + v + kh * 8;
      outp[(size_t)r * Hx + col] = acc[q][v];
    }
  }
}

// ---------------------------------------------------------------------------
// Kernel 2: copy x into comb[:, 0:64]  (comb is 4096 x 192 row-major)
// ---------------------------------------------------------------------------
__global__ void __launch_bounds__(256)
copyx_kernel(const float* __restrict__ x, float* __restrict__ comb) {
  int idx = blockIdx.x * blockDim.x + threadIdx.x; // 0..BN*Dx-1
  int r = idx >> 6;        // /64
  int c = idx & 63;
  comb[(size_t)r * (Dx + Hx) + c] = x[idx];
}

// ---------------------------------------------------------------------------
// Kernel 3: pairwise attention + message aggregation.
// One 128-thread block (4 wave32) per (b,i).
//   phase1: logits[j] = relu(PAi[i]+PAj[j]+ab1) . A2 + ab2  (wave32 reduce)
//   phase2: masked softmax over j (block reduce in LDS)
//   phase3: aggH[h] = sum_j attn[j]*relu(P1i[h]+P1j[j,h]+b1[h])  (coalesced)
// ---------------------------------------------------------------------------
__global__ void __launch_bounds__(128)
pair_kernel(const float* __restrict__ P, const int* __restrict__ masks,
            const float* __restrict__ b1, const float* __restrict__ ab1,
            const float* __restrict__ A2, const float* __restrict__ ab2,
            float* __restrict__ aggH, float* __restrict__ sAout) {
  const int b = blockIdx.y;
  const int i = blockIdx.x;
  const int t = threadIdx.x;      // 0..127
  const int lane = t & 31;
  const int wave = t >> 5;

  const float* P1i = P + 0 * (size_t)BN * Hx;
  const float* P1j = P + 1 * (size_t)BN * Hx;
  const float* PAi = P + 2 * (size_t)BN * Hx;
  const float* PAj = P + 3 * (size_t)BN * Hx;

  __shared__ __align__(16) float sPAi[Hx];
  __shared__ __align__(16) float sP1i[Hx];
  __shared__ float sL[Nx];
  __shared__ float sRed[Nx];

  const int bi = b * Nx + i;
  sPAi[t] = PAi[(size_t)bi * Hx + t];
  sP1i[t] = P1i[(size_t)bi * Hx + t];
  __syncthreads();

  // ---- phase 1: each wave computes 32 logits, lanes split H in float4s ----
  const float4 pai = ((const float4*)sPAi)[lane];
  const float4 ab1v = ((const float4*)ab1)[lane];
  const float4 a2v = ((const float4*)A2)[lane];
  const float ab2v = ab2[0];

  for (int jj = 0; jj < 32; ++jj) {
    const int j = wave * 32 + jj;
    const float* prow = PAj + ((size_t)(b * Nx + j)) * Hx;
    if (jj + 1 < 32)
      __builtin_prefetch(PAj + ((size_t)(b * Nx + j + 1)) * Hx + 4 * lane, 0, 3);
    const float4 paj = ((const float4*)prow)[lane];
    float p;
    p  = fmaxf(pai.x + paj.x + ab1v.x, 0.f) * a2v.x;
    p += fmaxf(pai.y + paj.y + ab1v.y, 0.f) * a2v.y;
    p += fmaxf(pai.z + paj.z + ab1v.z, 0.f) * a2v.z;
    p += fmaxf(pai.w + paj.w + ab1v.w, 0.f) * a2v.w;
    for (int off = 16; off; off >>= 1) p += __shfl_xor(p, off, 32);
    if (lane == 0)
      sL[j] = masks[b * Nx + j] ? (p + ab2v) : -__builtin_inff();
  }
  __syncthreads();

  // ---- phase 2: masked softmax over 128 logits ----
  const float v = sL[t];
  sRed[t] = v;
  __syncthreads();
  for (int s = 64; s; s >>= 1) {
    if (t < s) sRed[t] = fmaxf(sRed[t], sRed[t + s]);
    __syncthreads();
  }
  const float m = sRed[0];
  __syncthreads();
  const float e = (v == -__builtin_inff()) ? 0.f : __expf(v - m);
  sRed[t] = e;
  __syncthreads();
  for (int s = 64; s; s >>= 1) {
    if (t < s) sRed[t] += sRed[t + s];
    __syncthreads();
  }
  const float S = sRed[0];
  const float attn = (S > 0.f) ? (e / S) : 0.f;
  __syncthreads();
  sL[t] = attn;
  __syncthreads();

  // ---- phase 3: weighted aggregation of message hidden (coalesced in t) ----
  const float p1 = sP1i[t];
  const float b1t = b1[t];
  const float* pj = P1j + (size_t)(b * Nx) * Hx + t;
  float acc = 0.f;
#pragma unroll 4
  for (int j = 0; j < Nx; ++j) {
    const float h = p1 + pj[(size_t)j * Hx] + b1t;
    acc += sL[j] * fmaxf(h, 0.f);
  }
  aggH[(size_t)bi * Hx + t] = acc;
  if (t == 0) sAout[bi] = (S > 0.f) ? 1.f : 0.f;
}

// ---------------------------------------------------------------------------
// Kernel 4: aggM = aggH @ W2 + b2 * sA  -> comb[:, 64:192]
// grid 256 M-tiles, 8 waves/block, wave w owns N tile w.  K = 128.
// ---------------------------------------------------------------------------
__global__ void __launch_bounds__(256)
aggm_kernel(const float* __restrict__ aggH, const float* __restrict__ W2,
            const float* __restrict__ b2, const float* __restrict__ sA,
            float* __restrict__ comb) {
  const int lane = threadIdx.x & 31;
  const int wave = threadIdx.x >> 5;
  const int mbase = blockIdx.x * 16;
  const int row = lane & 15;
  const int kh = lane >> 4;
  const int col = wave * 16 + row;

  const float* arow = aggH + (size_t)(mbase + row) * Hx;
  v8f acc = {};
  for (int ks = 0; ks < Hx / 4; ++ks) {
    const int k0 = ks * 4 + kh * 2;
    v2f a;
    a.x = arow[k0];
    a.y = arow[k0 + 1];
    v2f bf;
    bf.x = W2[(size_t)k0 * Hx + col];
    bf.y = W2[(size_t)(k0 + 1) * Hx + col];
    acc = wmma4(a, bf, acc);
  }
  const float b2c = b2[col];
  for (int v = 0; v < 8; ++v) {
    const int r = mbase + v + kh * 8;
    comb[(size_t)r * (Dx + Hx) + Dx + col] = acc[v] + b2c * sA[r];
  }
}

// ---------------------------------------------------------------------------
// Kernel 5: Hup = relu(comb @ U1 + ub1)   (4096x192)@(192x128), K = 192.
// ---------------------------------------------------------------------------
__global__ void __launch_bounds__(256)
hup_kernel(const float* __restrict__ comb, const float* __restrict__ U1,
           const float* __restrict__ ub1, float* __restrict__ Hup) {
  const int lane = threadIdx.x & 31;
  const int wave = threadIdx.x >> 5;
  const int mbase = blockIdx.x * 16;
  const int row = lane & 15;
  const int kh = lane >> 4;
  const int col = wave * 16 + row;
  const int K = Dx + Hx; // 192

  const float* arow = comb + (size_t)(mbase + row) * K;
  v8f acc = {};
  for (int ks = 0; ks < K / 4; ++ks) {
    const int k0 = ks * 4 + kh * 2;
    v2f a;
    a.x = arow[k0];
    a.y = arow[k0 + 1];
    v2f bf;
    bf.x = U1[(size_t)k0 * Hx + col];
    bf.y = U1[(size_t)(k0 + 1) * Hx + col];
    acc = wmma4(a, bf, acc);
  }
  const float bc = ub1[col];
  for (int v = 0; v < 8; ++v) {
    const int r = mbase + v + kh * 8;
    Hup[(size_t)r * Hx + col] = fmaxf(acc[v] + bc, 0.f);
  }
}

// ---------------------------------------------------------------------------
// Kernel 6: out = x + Hup @ U2 + ub2   (4096x128)@(128x64), 4 N tiles.
// ---------------------------------------------------------------------------
__global__ void __launch_bounds__(128)
out_kernel(const float* __restrict__ Hup, const float* __restrict__ U2,
           const float* __restrict__ ub2, const float* __restrict__ x,
           float* __restrict__ out) {
  const int lane = threadIdx.x & 31;
  const int wave = threadIdx.x >> 5; // 0..3 -> N tile
  const int mbase = blockIdx.x * 16;
  const int row = lane & 15;
  const int kh = lane >> 4;
  const int col = wave * 16 + row;

  const float* arow = Hup + (size_t)(mbase + row) * Hx;
  v8f acc = {};
  for (int ks = 0; ks < Hx / 4; ++ks) {
    const int k0 = ks * 4 + kh * 2;
    v2f a;
    a.x = arow[k0];
    a.y = arow[k0 + 1];
    v2f bf;
    bf.x = U2[(size_t)k0 * Dx + col];
    bf.y = U2[(size_t)(k0 + 1) * Dx + col];
    acc = wmma4(a, bf, acc);
  }
  const float bc = ub2[col];
  for (int v = 0; v < 8; ++v) {
    const int r = mbase + v + kh * 8;
    out[(size_t)r * Dx + col] = x[(size_t)r * Dx + col] + acc[v] + bc;
  }
}

// ---------------------------------------------------------------------------
extern "C" void kernel_launch(void* const* d_in, const int* in_sizes, int n_in,
                              void* d_out, int out_size, void* d_ws,
                              size_t ws_size, hipStream_t stream) {
  const float* x = (const float*)d_in[0];
  const int* masks = (const int*)d_in[1];
  const float* W1 = (const float*)d_in[2];
  const float* b1 = (const float*)d_in[3];
  const float* W2 = (const float*)d_in[4];
  const float* b2 = (const float*)d_in[5];
  const float* A1 = (const float*)d_in[6];
  const float* ab1 = (const float*)d_in[7];
  const float* A2 = (const float*)d_in[8];
  const float* ab2 = (const float*)d_in[9];
  const float* U1 = (const float*)d_in[10];
  const float* ub1 = (const float*)d_in[11];
  const float* U2 = (const float*)d_in[12];
  const float* ub2 = (const float*)d_in[13];
  float* out = (float*)d_out;

  // workspace layout (floats): 4 projections, aggH, sA, comb, Hup  (~15 MB)
  float* P = (float*)d_ws;
  float* aggH = P + (size_t)4 * BN * Hx;
  float* sA = aggH + (size_t)BN * Hx;
  float* comb = sA + BN;
  float* Hup = comb + (size_t)BN * (Dx + Hx);

  proj_kernel<<<256, 256, 0, stream>>>(x, W1, A1, P);
  copyx_kernel<<<(BN * Dx) / 256, 256, 0, stream>>>(x, comb);
  pair_kernel<<<dim3(Nx, Bx), 128, 0, stream>>>(P, masks, b1, ab1, A2, ab2,
                                                aggH, sA);
  aggm_kernel<<<256, 256, 0, stream>>>(aggH, W2, b2, sA, comb);
  hup_kernel<<<256, 256, 0, stream>>>(comb, U1, ub1, Hup);
  out_kernel<<<256, 128, 0, stream>>>(Hup, U2, ub2, x, out);
}